// DeltaNet_31877247271478
// MI455X (gfx1250) — compile-verified
//
#include <hip/hip_runtime.h>
#include <hip/hip_bf16.h>

typedef __attribute__((ext_vector_type(2))) float v2f;
typedef __attribute__((ext_vector_type(8))) float v8f;

constexpr int D    = 64;    // head dim (dk == dv)
constexpr int Cc   = 32;    // chunk size
constexpr int Lseq = 8192;  // sequence length
constexpr int NCH  = Lseq / Cc;   // 256 chunks
constexpr int BH   = 2 * 16;      // b*h = 32

// f32 WMMA 16x16x4: D = A(16x4) * B(4x16) + C
__device__ __forceinline__ v8f wmma4(v2f a, v2f b, v8f c) {
  return __builtin_amdgcn_wmma_f32_16x16x4_f32(
      /*neg_a=*/false, a, /*neg_b=*/false, b,
      /*c_mod=*/(short)0, c, /*reuse_a=*/false, /*reuse_b=*/false);
}

// Generic pointer to a __shared__ object: low 32 bits are the LDS byte offset
// (flat LDS aperture truncates addr[31:0], same as addrspacecast generic->local).
__device__ __forceinline__ unsigned ldsoff(const void* p) {
  return (unsigned)(size_t)p;
}

// CDNA5 async copy global->LDS, 16B per lane, tracked by ASYNCcnt (GVS mode:
// global address = SGPR base + 32-bit VGPR byte offset + imm offset; VDST holds the LDS address).
__device__ __forceinline__ void async_b128(unsigned lds, unsigned goff,
                                           const float* base) {
  asm volatile("global_load_async_to_lds_b128 %0, %1, %2 offset:0"
               :: "v"(lds), "v"(goff), "s"(base)
               : "memory");
}

// ---------------------------------------------------------------------------
// Kernel A: l2-normalize q,k over d=64; v*beta; k_beta = kn*beta.
// One wave (32 lanes) per token row, 2 floats per lane.
// ---------------------------------------------------------------------------
__global__ __launch_bounds__(256) void deltanet_norm(
    const float* __restrict__ q, const float* __restrict__ k,
    const float* __restrict__ v, const float* __restrict__ beta,
    float* __restrict__ qn, float* __restrict__ kn,
    float* __restrict__ vb, float* __restrict__ kb)
{
  const int tid  = threadIdx.x;
  const int wave = tid >> 5;
  const int lane = tid & 31;
  const long row = (long)blockIdx.x * 8 + wave;   // token index over b*h*L
  const size_t off = (size_t)row * D + lane * 2;

  float2 q2 = *(const float2*)(q + off);
  float2 k2 = *(const float2*)(k + off);
  float2 v2 = *(const float2*)(v + off);
  const float bta = beta[row];

  float qs = q2.x * q2.x + q2.y * q2.y;
  float ks = k2.x * k2.x + k2.y * k2.y;
#pragma unroll
  for (int m = 16; m; m >>= 1) {
    qs += __shfl_xor(qs, m, 32);
    ks += __shfl_xor(ks, m, 32);
  }
  const float qr = rsqrtf(qs + 1e-6f);
  const float kr = rsqrtf(ks + 1e-6f);

  float2 qo = {q2.x * qr, q2.y * qr};
  float2 ko = {k2.x * kr, k2.y * kr};
  float2 vo = {v2.x * bta, v2.y * bta};
  float2 kbo = {ko.x * bta, ko.y * bta};
  *(float2*)(qn + off) = qo;
  *(float2*)(kn + off) = ko;
  *(float2*)(vb + off) = vo;
  *(float2*)(kb + off) = kbo;
}

// ---------------------------------------------------------------------------
// Kernel B: per chunk: G = k_beta @ kn^T (WMMA, 2x2 tiles of 16x16, K=64),
// then forward substitution of unit-lower M = I + tril(G,-1) against v_beta
// (-> u) and k_beta (-> w), written in place over the vb/kb buffers.
// 128 threads = 4 waves; one WMMA tile per wave; one RHS column per thread.
// ---------------------------------------------------------------------------
__global__ __launch_bounds__(128) void deltanet_chunk(
    const float* __restrict__ knb, float* uvb, float* wkb)
{
  __shared__ float sKn[Cc * D];
  __shared__ float sKb[Cc * D];
  __shared__ float sVb[Cc * D];
  __shared__ float sG[Cc * Cc];

  const int tid  = threadIdx.x;
  const int wave = tid >> 5;
  const int lane = tid & 31;
  const int half = lane >> 4;
  const int l15  = lane & 15;

  const int chunk = blockIdx.x & (NCH - 1);
  const int bh    = blockIdx.x >> 8;   // NCH == 256
  const size_t cbase = ((size_t)bh * Lseq + (size_t)chunk * Cc) * D;

  for (int i = tid; i < (Cc * D) / 4; i += 128) {
    ((float4*)sKn)[i] = ((const float4*)(knb + cbase))[i];
    ((float4*)sVb)[i] = ((const float4*)(uvb + cbase))[i];
    ((float4*)sKb)[i] = ((const float4*)(wkb + cbase))[i];
  }
  __syncthreads();

  // G = kb @ kn^T   (B operand is transposed -> row-major fetch of kn)
  {
    const int ti = wave >> 1, tj = wave & 1;
    v8f acc = {};
#pragma unroll
    for (int k0 = 0; k0 < D; k0 += 4) {
      const int kk = k0 + 2 * half;
      v2f a, b;
      a.x = sKb[(ti * 16 + l15) * D + kk];
      a.y = sKb[(ti * 16 + l15) * D + kk + 1];
      b.x = sKn[(tj * 16 + l15) * D + kk];
      b.y = sKn[(tj * 16 + l15) * D + kk + 1];
      acc = wmma4(a, b, acc);
    }
#pragma unroll
    for (int r = 0; r < 8; ++r)
      sG[(ti * 16 + r + 8 * half) * Cc + tj * 16 + l15] = acc[r];
  }
  __syncthreads();

  // Forward substitution: x_i = rhs_i - sum_{j<i} G[i][j] x_j
  {
    const int col = tid & 63;
    const int sel = tid >> 6;            // 0 -> u (from vb), 1 -> w (from kb)
    const float* src = sel ? sKb : sVb;
    float* dst = (sel ? wkb : uvb) + cbase;
    float x[Cc];
#pragma unroll
    for (int i = 0; i < Cc; ++i) {
      float acc = src[i * D + col];
#pragma unroll
      for (int j = 0; j < i; ++j) acc -= sG[i * Cc + j] * x[j];
      x[i] = acc;
    }
#pragma unroll
    for (int i = 0; i < Cc; ++i) dst[i * D + col] = x[i];
  }
}

// ---------------------------------------------------------------------------
// Kernel C: sequential inter-chunk scan. One workgroup per (b,h); S (64x64)
// lives in LDS for all 256 chunk steps. Inputs for chunk ch+1 are prefetched
// with global_load_async_to_lds_b128 (double buffered, ASYNCcnt-tracked)
// while chunk ch computes on v_wmma_f32_16x16x4_f32.
// 256 threads = 8 waves; each wave owns 16x16 output tiles.
// ---------------------------------------------------------------------------
__global__ __launch_bounds__(256) void deltanet_scan(
    const float* __restrict__ qn, const float* __restrict__ kn,
    const float* __restrict__ ub, const float* __restrict__ wb,
    float* __restrict__ o, float* __restrict__ Sout)
{
  __shared__ float sQ [2][Cc * D];
  __shared__ float sK [2][Cc * D];
  __shared__ float sU0[2][Cc * D];
  __shared__ float sW [2][Cc * D];
  __shared__ float sU[Cc * D];
  __shared__ float sP[Cc * Cc];
  __shared__ float sS[D * D];

  const int tid  = threadIdx.x;
  const int wave = tid >> 5;
  const int lane = tid & 31;
  const int half = lane >> 4;
  const int l15  = lane & 15;

  const int bh = blockIdx.x;
  const size_t base = (size_t)bh * Lseq * D;

  for (int i = tid; i < D * D; i += 256) sS[i] = 0.f;

  // Issue one chunk's worth of async copies (4 buffers x 2 float4 per thread
  // = 8 ASYNCcnt increments per wave).
  auto issue = [&](int buf, int ch) {
#pragma unroll
    for (int s = 0; s < 2; ++s) {
      const int idx = tid + s * 256;                       // float4 index 0..511
      const unsigned go =
          (unsigned)((base + (size_t)ch * Cc * D + (size_t)idx * 4) * sizeof(float));
      async_b128(ldsoff(&sQ [buf][idx * 4]), go, qn);
      async_b128(ldsoff(&sK [buf][idx * 4]), go, kn);
      async_b128(ldsoff(&sU0[buf][idx * 4]), go, ub);
      async_b128(ldsoff(&sW [buf][idx * 4]), go, wb);
    }
  };

  issue(0, 0);

  for (int ch = 0; ch < NCH; ++ch) {
    const int par = ch & 1;
    if (ch + 1 < NCH) {
      issue(par ^ 1, ch + 1);                   // prefetch next chunk
      // 8 newly-issued may remain in flight; in-order retirement guarantees
      // the previous batch (current chunk's data) has landed.
      asm volatile("s_wait_asynccnt 0x8" ::: "memory");
    } else {
      asm volatile("s_wait_asynccnt 0x0" ::: "memory");
    }
    __syncthreads();

    const float* cQ  = sQ [par];
    const float* cK  = sK [par];
    const float* cU0 = sU0[par];
    const float* cW  = sW [par];
    const size_t cbase = base + (size_t)ch * Cc * D;

    // Phase 1: U = U0 - W @ S  (8 tiles) ; waves 0-3 also P = tril(Q @ K^T)
    {
      const int ti = wave >> 2, tj = wave & 3;
      v8f acc = {};
#pragma unroll
      for (int k0 = 0; k0 < D; k0 += 4) {
        const int kk = k0 + 2 * half;
        v2f a, b;
        a.x = cW[(ti * 16 + l15) * D + kk];
        a.y = cW[(ti * 16 + l15) * D + kk + 1];
        b.x = sS[kk * D + tj * 16 + l15];
        b.y = sS[(kk + 1) * D + tj * 16 + l15];
        acc = wmma4(a, b, acc);
      }
#pragma unroll
      for (int r = 0; r < 8; ++r) {
        const int row = ti * 16 + r + 8 * half;
        const int col = tj * 16 + l15;
        sU[row * D + col] = cU0[row * D + col] - acc[r];
      }
      if (wave < 4) {
        const int pi = wave >> 1, pj = wave & 1;
        v8f pc = {};
#pragma unroll
        for (int k0 = 0; k0 < D; k0 += 4) {
          const int kk = k0 + 2 * half;
          v2f a, b;
          a.x = cQ[(pi * 16 + l15) * D + kk];
          a.y = cQ[(pi * 16 + l15) * D + kk + 1];
          b.x = cK[(pj * 16 + l15) * D + kk];
          b.y = cK[(pj * 16 + l15) * D + kk + 1];
          pc = wmma4(a, b, pc);
        }
#pragma unroll
        for (int r = 0; r < 8; ++r) {
          const int row = pi * 16 + r + 8 * half;
          const int col = pj * 16 + l15;
          sP[row * Cc + col] = (row >= col) ? pc[r] : 0.f;
        }
      }
    }
    __syncthreads();

    // Phase 2: O = Q @ S + P @ U ; stream to global (non-temporal, write-once)
    {
      const int ti = wave >> 2, tj = wave & 3;
      v8f acc = {};
#pragma unroll
      for (int k0 = 0; k0 < D; k0 += 4) {
        const int kk = k0 + 2 * half;
        v2f a, b;
        a.x = cQ[(ti * 16 + l15) * D + kk];
        a.y = cQ[(ti * 16 + l15) * D + kk + 1];
        b.x = sS[kk * D + tj * 16 + l15];
        b.y = sS[(kk + 1) * D + tj * 16 + l15];
        acc = wmma4(a, b, acc);
      }
#pragma unroll
      for (int k0 = 0; k0 < Cc; k0 += 4) {
        const int kk = k0 + 2 * half;
        v2f a, b;
        a.x = sP[(ti * 16 + l15) * Cc + kk];
        a.y = sP[(ti * 16 + l15) * Cc + kk + 1];
        b.x = sU[kk * D + tj * 16 + l15];
        b.y = sU[(kk + 1) * D + tj * 16 + l15];
        acc = wmma4(a, b, acc);
      }
#pragma unroll
      for (int r = 0; r < 8; ++r) {
        const int row = ti * 16 + r + 8 * half;
        const int col = tj * 16 + l15;
        __builtin_nontemporal_store(acc[r], &o[cbase + row * D + col]);
      }
    }
    __syncthreads();   // all reads of S complete before update

    // Phase 3: S += K^T @ U  (16 tiles; wave w owns tiles 2w, 2w+1)
#pragma unroll
    for (int s = 0; s < 2; ++s) {
      const int t  = wave * 2 + s;
      const int ti = t >> 2, tj = t & 3;
      v8f acc;
#pragma unroll
      for (int r = 0; r < 8; ++r)
        acc[r] = sS[(ti * 16 + r + 8 * half) * D + tj * 16 + l15];
#pragma unroll
      for (int k0 = 0; k0 < Cc; k0 += 4) {
        const int kk = k0 + 2 * half;
        v2f a, b;
        a.x = cK[kk * D + ti * 16 + l15];        // (K^T)[m][k] = K[k][m]
        a.y = cK[(kk + 1) * D + ti * 16 + l15];
        b.x = sU[kk * D + tj * 16 + l15];
        b.y = sU[(kk + 1) * D + tj * 16 + l15];
        acc = wmma4(a, b, acc);
      }
#pragma unroll
      for (int r = 0; r < 8; ++r)
        sS[(ti * 16 + r + 8 * half) * D + tj * 16 + l15] = acc[r];
    }
    __syncthreads();
  }

  float* So = Sout + (size_t)bh * D * D;
  for (int i = tid; i < D * D; i += 256) So[i] = sS[i];
}

// ---------------------------------------------------------------------------
extern "C" void kernel_launch(void* const* d_in, const int* in_sizes, int n_in,
                              void* d_out, int out_size, void* d_ws, size_t ws_size,
                              hipStream_t stream) {
  const float* q    = (const float*)d_in[0];
  const float* k    = (const float*)d_in[1];
  const float* v    = (const float*)d_in[2];
  const float* beta = (const float*)d_in[3];
  // d_in[4] = chunk_size (32), compile-time constant here.

  float* out = (float*)d_out;
  float* ws  = (float*)d_ws;

  const size_t TOKD = (size_t)BH * Lseq * D;   // 16,777,216 floats per tensor
  float* qn  = ws;
  float* knb = ws + TOKD;
  float* uv  = ws + 2 * TOKD;   // v*beta -> u (in place)
  float* wk  = ws + 3 * TOKD;   // k_beta -> w (in place)

  deltanet_norm<<<(BH * Lseq) / 8, 256, 0, stream>>>(q, k, v, beta, qn, knb, uv, wk);
  deltanet_chunk<<<BH * NCH, 128, 0, stream>>>(knb, uv, wk);
  deltanet_scan<<<BH, 256, 0, stream>>>(qn, knb, uv, wk, out, out + TOKD);
}